// DLP_Loss_19610820673960
// MI455X (gfx1250) — compile-verified
//
#include <hip/hip_runtime.h>
#include <hip/hip_bf16.h>
#include <math.h>

// Problem constants (from reference)
#define NN   8192
#define DD   128
#define CC   7
#define KK   5
#define EPSV 1e-6f
#define NTILES (NN / 16)     // 512 column tiles of 16
#define BIGF 3.0e38f

typedef __attribute__((ext_vector_type(2))) float v2f;
typedef __attribute__((ext_vector_type(8))) float v8f;

// ---------------------------------------------------------------- init out
__global__ void dlp_init_kernel(float* out) {
    if (threadIdx.x == 0 && blockIdx.x == 0) out[0] = 0.0f;
}

// ---------------------------------------------------------------- row norms
// ra[i] = sum((x_i + eps)^2), rb[i] = sum(x_i^2)
__global__ __launch_bounds__(256)
void dlp_norms_kernel(const float* __restrict__ x,
                      float* __restrict__ ra, float* __restrict__ rb) {
    int i = blockIdx.x * blockDim.x + threadIdx.x;
    if (i >= NN) return;
    const float4* xp = (const float4*)(x + (size_t)i * DD);
    float sa = 0.0f, sb = 0.0f;
#pragma unroll
    for (int q = 0; q < DD / 4; ++q) {
        float4 v = xp[q];
        float a0 = v.x + EPSV, a1 = v.y + EPSV, a2 = v.z + EPSV, a3 = v.w + EPSV;
        sa += a0 * a0 + a1 * a1 + a2 * a2 + a3 * a3;
        sb += v.x * v.x + v.y * v.y + v.z * v.z + v.w * v.w;
    }
    ra[i] = sa;
    rb[i] = sb;
}

// ---------------------------------------------------------------- cross-entropy
__global__ __launch_bounds__(256)
void dlp_ce_kernel(const float* __restrict__ scores,
                   const int* __restrict__ target,
                   float* __restrict__ out) {
    int i = blockIdx.x * blockDim.x + threadIdx.x;
    if (i >= NN) return;
    const float* s = scores + (size_t)i * CC;
    float m = s[0];
#pragma unroll
    for (int c = 1; c < CC; ++c) m = fmaxf(m, s[c]);
    float sum = 0.0f;
#pragma unroll
    for (int c = 0; c < CC; ++c) sum += __expf(s[c] - m);
    float lse = m + __logf(sum);
    float ce  = lse - s[target[i]];
    atomicAdd(out, ce * (1.0f / (float)NN));
}

// ---------------------------------------------------------------- sorted top-5 insert (registers only)
#define TOP5_INSERT(v, j)                                                        \
    do {                                                                         \
        float _v = (v); int _j = (j);                                            \
        if (_v < b4) {                                                           \
            if (_v < b3) { b4 = b3; i4 = i3;                                     \
                if (_v < b2) { b3 = b2; i3 = i2;                                 \
                    if (_v < b1) { b2 = b1; i2 = i1;                             \
                        if (_v < b0) { b1 = b0; i1 = i0; b0 = _v; i0 = _j; }     \
                        else { b1 = _v; i1 = _j; }                               \
                    } else { b2 = _v; i2 = _j; }                                 \
                } else { b3 = _v; i3 = _j; }                                     \
            } else { b4 = _v; i4 = _j; }                                         \
        }                                                                        \
    } while (0)

// ---------------------------------------------------------------- fused distance GEMM + masked top-K + pair loss
// One workgroup (4 wave32) per 16-row tile. Wave w handles column tiles w, w+4, ...
__global__ __launch_bounds__(128)
void dlp_tile_kernel(const float* __restrict__ x,
                     const int* __restrict__ tgt,
                     const float* __restrict__ ra,
                     const float* __restrict__ rb,
                     float* __restrict__ out) {
    const int lane = threadIdx.x & 31;
    const int lh   = lane >> 4;          // half-wave: 0 or 1
    const int l16  = lane & 15;
    // wave id: wave-uniform -> force scalar so loop control stays on SALU and
    // EXEC is untouched at the WMMAs (ISA requires EXEC all-1s for WMMA).
    const int wv   = __builtin_amdgcn_readfirstlane(threadIdx.x >> 5);  // 0..3
    const int row_base = blockIdx.x * 16;

    __shared__ float tile[4][16][17];    // per-wave 16x16 d2 tile (padded)
    __shared__ float mVal[16][4][KK];    // per-row per-wave top-5 values
    __shared__ int   mIdx[16][4][KK];
    __shared__ float fVal[16][KK];       // merged final top-5
    __shared__ int   fIdx[16][KK];

    // ---- preload A fragments for this 16-row block: 32 chunks of K=4.
    // A 16x4 f32 layout: lane%16 = M, v0/v1 hold k = 4*kc + 2*lh + {0,1}
    v2f afrag[DD / 4];
    {
        const float* ap = x + (size_t)(row_base + l16) * DD + 2 * lh;
#pragma unroll
        for (int kc = 0; kc < DD / 4; ++kc) {
            float2 t = *(const float2*)(ap + 4 * kc);
            v2f a; a.x = t.x + EPSV; a.y = t.y + EPSV;
            afrag[kc] = a;
        }
    }
    // ---- per-C-VGPR row metadata: C layout row = r + 8*lh
    float raR[8]; int tgR[8]; int rowR[8];
#pragma unroll
    for (int r = 0; r < 8; ++r) {
        int row = row_base + r + 8 * lh;
        rowR[r] = row;
        raR[r]  = ra[row];
        tgR[r]  = tgt[row];
    }

    // owner-lane (lanes 0..15 of each wave) running top-5, sorted ascending
    float b0 = BIGF, b1 = BIGF, b2 = BIGF, b3 = BIGF, b4 = BIGF;
    int   i0 = -1,  i1 = -1,  i2 = -1,  i3 = -1,  i4 = -1;

    for (int ct = wv; ct < NTILES; ct += 4) {
        const int colg = ct * 16 + l16;
        const float* bp = x + (size_t)colg * DD + 2 * lh;

        // ---- batch-load the full B tile into registers first so all 32
        // b64 loads are in flight together (one L2 latency per tile, not 32)
        v2f bfrag[DD / 4];
#pragma unroll
        for (int kc = 0; kc < DD / 4; ++kc) {
            float2 t = *(const float2*)(bp + 4 * kc);
            v2f b; b.x = t.x; b.y = t.y;
            bfrag[kc] = b;
        }
        const float rbv = rb[colg];     // overlap with WMMA burst
        const int   tgc = tgt[colg];

        // ---- dense WMMA burst, dual accumulators break the C RAW chain
        v8f acc0 = {}, acc1 = {};
#pragma unroll
        for (int kc = 0; kc < DD / 4; kc += 2) {
            acc0 = __builtin_amdgcn_wmma_f32_16x16x4_f32(
                false, afrag[kc], false, bfrag[kc], (short)0, acc0, false, false);
            acc1 = __builtin_amdgcn_wmma_f32_16x16x4_f32(
                false, afrag[kc + 1], false, bfrag[kc + 1], (short)0, acc1, false, false);
        }

#pragma unroll
        for (int r = 0; r < 8; ++r) {
            float s  = acc0[r] + acc1[r];
            float d2 = raR[r] - 2.0f * s + rbv;
            bool valid = (tgR[r] == tgc) && (rowR[r] != colg);
            tile[wv][r + 8 * lh][l16] = valid ? d2 : BIGF;
        }
        asm volatile("s_wait_dscnt 0" ::: "memory");  // cross-lane LDS RAW inside wave

        if (lh == 0) {  // lane l16 owns row l16: scan this tile's 16 columns
#pragma unroll
            for (int cc = 0; cc < 16; ++cc) {
                float v = tile[wv][l16][cc];
                TOP5_INSERT(v, ct * 16 + cc);
            }
        }
    }

    // ---- stash per-wave top-5
    if (lh == 0) {
        mVal[l16][wv][0] = b0; mIdx[l16][wv][0] = i0;
        mVal[l16][wv][1] = b1; mIdx[l16][wv][1] = i1;
        mVal[l16][wv][2] = b2; mIdx[l16][wv][2] = i2;
        mVal[l16][wv][3] = b3; mIdx[l16][wv][3] = i3;
        mVal[l16][wv][4] = b4; mIdx[l16][wv][4] = i4;
    }
    __syncthreads();

    // ---- wave 0 merges 4x5 candidates -> final top-5 per row
    if (wv == 0 && lh == 0) {
        b0 = b1 = b2 = b3 = b4 = BIGF;
        i0 = i1 = i2 = i3 = i4 = -1;
#pragma unroll
        for (int ww = 0; ww < 4; ++ww) {
#pragma unroll
            for (int k = 0; k < KK; ++k) {
                TOP5_INSERT(mVal[l16][ww][k], mIdx[l16][ww][k]);
            }
        }
        fVal[l16][0] = b0; fIdx[l16][0] = i0;
        fVal[l16][1] = b1; fIdx[l16][1] = i1;
        fVal[l16][2] = b2; fIdx[l16][2] = i2;
        fVal[l16][3] = b3; fIdx[l16][3] = i3;
        fVal[l16][4] = b4; fIdx[l16][4] = i4;
    }
    __syncthreads();

    // ---- exact mse recompute (reference recomputes from gathered x, no eps)
    int p = threadIdx.x;
    if (p < 16 * KK) {
        int rloc = p / KK, k = p % KK;
        float v = fVal[rloc][k];
        int   j = fIdx[rloc][k];
        if (v < 1.0e37f && j >= 0) {   // vmask = isfinite
            const float4* xi = (const float4*)(x + (size_t)(row_base + rloc) * DD);
            const float4* xj = (const float4*)(x + (size_t)j * DD);
            float s = 0.0f;
#pragma unroll
            for (int q = 0; q < DD / 4; ++q) {
                float4 a = xi[q], bb = xj[q];
                float d0 = a.x - bb.x, d1 = a.y - bb.y, d2 = a.z - bb.z, d3 = a.w - bb.w;
                s += d0 * d0 + d1 * d1 + d2 * d2 + d3 * d3;
            }
            float mse = s * (1.0f / (float)DD);
            atomicAdd(out, mse * (0.5f / (float)KK));
        }
    }
}

// ---------------------------------------------------------------- launcher
extern "C" void kernel_launch(void* const* d_in, const int* in_sizes, int n_in,
                              void* d_out, int out_size, void* d_ws, size_t ws_size,
                              hipStream_t stream) {
    const float* x      = (const float*)d_in[0];  // (N, D) fp32
    const float* scores = (const float*)d_in[1];  // (N, C) fp32
    const int*   target = (const int*)d_in[2];    // (N,)   int32
    float* out = (float*)d_out;                   // scalar

    float* ra = (float*)d_ws;                     // N floats
    float* rb = ra + NN;                          // N floats

    dlp_init_kernel<<<1, 32, 0, stream>>>(out);
    dlp_norms_kernel<<<(NN + 255) / 256, 256, 0, stream>>>(x, ra, rb);
    dlp_ce_kernel<<<(NN + 255) / 256, 256, 0, stream>>>(scores, target, out);
    dlp_tile_kernel<<<NN / 16, 128, 0, stream>>>(x, target, ra, rb, out);
}